// Imputer_56341380989407
// MI455X (gfx1250) — compile-verified
//
#include <hip/hip_runtime.h>

// Imputer(type='MEAN') for x[B=64, D=8, N=2048, L=64] f32, missing == -inf.
// Memory-bound: 512 MiB traffic -> ~22us floor @ 23.3 TB/s. Strategy:
// one wave32 per (b,n); each lane holds 2 elems of each of the 8 rows
// (16 VGPRs of payload), ballot_w32 finds first-missing d (uniform),
// shfl_xor tree reduces observed sum/count, NT loads/stores stream HBM.

typedef float v2f __attribute__((ext_vector_type(2)));

namespace {
constexpr int kB = 64;
constexpr int kD = 8;
constexpr int kN = 2048;
constexpr int kL = 64;
constexpr int kWavesPerBlock = 8;           // 256 threads = 8 wave32
}

__global__ __launch_bounds__(256) void imputer_mean_kernel(
    const float* __restrict__ x, float* __restrict__ out) {
  const int lane  = threadIdx.x & 31;
  const int wslot = threadIdx.x >> 5;
  const unsigned wid = blockIdx.x * kWavesPerBlock + wslot;  // one wave per (b,n)
  const int n = wid & (kN - 1);       // N = 2^11
  const int b = wid >> 11;

  const size_t dStride = (size_t)kN * kL;                    // stride between d-rows
  const size_t base = ((size_t)b * kD) * dStride + (size_t)n * kL + (size_t)(lane * 2);

  // ---- Load all 8 rows for this (b,n): 8 coalesced 256B row reads, NT hint.
  v2f v[kD];
#pragma unroll
  for (int d = 0; d < kD; ++d) {
    v[d] = __builtin_nontemporal_load(
        (const v2f*)(x + base + (size_t)d * dStride));
  }

  const float NEG_INF = -__builtin_huge_valf();

  // ---- Find first d with any missing entry (ballot is wave-uniform), and
  //      capture this lane's observed-sum/count contribution from that row.
  bool  found = false;
  float s = 0.0f;
  float c = 0.0f;
#pragma unroll
  for (int d = 0; d < kD; ++d) {
    const bool m0 = (v[d].x == NEG_INF);
    const bool m1 = (v[d].y == NEG_INF);
    const unsigned bal = __builtin_amdgcn_ballot_w32(m0 | m1);
    if (!found && bal != 0u) {        // uniform branch: bal identical on all lanes
      found = true;
      s = (m0 ? 0.0f : v[d].x) + (m1 ? 0.0f : v[d].y);
      c = (m0 ? 0.0f : 1.0f) + (m1 ? 0.0f : 1.0f);
    }
  }

  // ---- Wave32 tree reduction (ds-swizzle path); broadcast mean to all lanes.
  float mean = 0.0f;
  if (found) {
#pragma unroll
    for (int off = 16; off > 0; off >>= 1) {
      s += __shfl_xor(s, off, 32);
      c += __shfl_xor(c, off, 32);
    }
    mean = (c > 0.0f) ? (s / c) : 0.0f;   // cnt==0 -> 0, matching reference
  }

  // ---- Rewrite all 8 rows, replacing -inf with mean; NT streaming stores.
#pragma unroll
  for (int d = 0; d < kD; ++d) {
    v2f w;
    w.x = (v[d].x == NEG_INF) ? mean : v[d].x;
    w.y = (v[d].y == NEG_INF) ? mean : v[d].y;
    __builtin_nontemporal_store(w, (v2f*)(out + base + (size_t)d * dStride));
  }
}

extern "C" void kernel_launch(void* const* d_in, const int* in_sizes, int n_in,
                              void* d_out, int out_size, void* d_ws, size_t ws_size,
                              hipStream_t stream) {
  (void)in_sizes; (void)n_in; (void)d_ws; (void)ws_size; (void)out_size;
  const float* x = (const float*)d_in[0];
  float* out = (float*)d_out;

  const int totalWaves = kB * kN;                       // 131072 (b,n) pairs
  const int blocks = totalWaves / kWavesPerBlock;       // 16384 blocks x 256 thr
  imputer_mean_kernel<<<blocks, 256, 0, stream>>>(x, out);
}